// SimGCL_Encoder_6347961663511
// MI455X (gfx1250) — compile-verified
//
#include <hip/hip_runtime.h>

#define USER_NUM    100000
#define ITEM_NUM    50000
#define CONCEPT_NUM 2000
#define NTOTAL      (USER_NUM + ITEM_NUM)
#define EMB         64
#define N_LAYERS    3
#define EPS_F       0.1f

typedef __attribute__((ext_vector_type(2))) float v2f;
typedef __attribute__((ext_vector_type(8))) float v8f;

// ---------------- utility kernels (grid-stride, float4) ----------------

__global__ void k_zero4(float4* __restrict__ p, int n4) {
  int i = blockIdx.x * blockDim.x + threadIdx.x;
  int stride = gridDim.x * blockDim.x;
  float4 z = {0.f, 0.f, 0.f, 0.f};
  for (; i < n4; i += stride) p[i] = z;
}

__global__ void k_copy4(float4* __restrict__ dst, const float4* __restrict__ src, int n4) {
  int i = blockIdx.x * blockDim.x + threadIdx.x;
  int stride = gridDim.x * blockDim.x;
  for (; i < n4; i += stride) dst[i] = src[i];
}

__global__ void k_scale4(float4* __restrict__ p, float s, int n4) {
  int i = blockIdx.x * blockDim.x + threadIdx.x;
  int stride = gridDim.x * blockDim.x;
  for (; i < n4; i += stride) {
    float4 v = p[i];
    v.x *= s; v.y *= s; v.z *= s; v.w *= s;
    p[i] = v;
  }
}

// ---------------- scatter SpMM: y[rows[e]+off] += scale*vals[e]*x[cols[e]] ----------------
// 16 lanes per edge, float4 per lane (covers 64 dims). State arrays (38.4 MB) are resident
// in the MI455X's 192 MB L2, so gathers and f32 atomics resolve at L2 bandwidth.

__global__ void k_scatter(const int* __restrict__ rows, const int* __restrict__ cols,
                          const float* __restrict__ vals, const float* __restrict__ x,
                          float* __restrict__ y, int nnz, float scale, int rowoff) {
  unsigned t = blockIdx.x * blockDim.x + threadIdx.x;
  unsigned e = t >> 4;
  if (e >= (unsigned)nnz) return;
  unsigned part = t & 15u;
  int r = rows[e] + rowoff;
  int c = cols[e];
  float v = vals[e] * scale;
  float4 xv = ((const float4*)(x + (size_t)c * EMB))[part];
  float* yp = y + (size_t)r * EMB + (size_t)part * 4;
  atomicAdd(yp + 0, v * xv.x);
  atomicAdd(yp + 1, v * xv.y);
  atomicAdd(yp + 2, v * xv.z);
  atomicAdd(yp + 3, v * xv.w);
}

// ---------------- per-row L2 norm of noise via V_WMMA_F32_16X16X4_F32 ----------------
// One wave handles 16 rows. A = squared 16x4 noise tile (ISA layout: lanes0-15 K=0,1;
// lanes16-31 K=2,3), B = ones(4x16)  =>  D[m][n] = sum_k A[m][k].  Accumulate 16 WMMAs
// to reduce all 64 dims. fac[row] = EPS / max(sqrt(sumsq), 1e-12).
// Launched as exactly nrows/16 waves (blockDim = 160 = 5 waves) so every wave that
// reaches the WMMA region has EXEC all-ones, as the ISA requires.

__global__ void k_norm_wmma(const float* __restrict__ nk, float* __restrict__ fac, int nrows) {
  int wave = (int)((blockIdx.x * blockDim.x + threadIdx.x) >> 5);
  int lane = threadIdx.x & 31;
  int base = wave * 16;
  if (base >= nrows) return;  // never taken with exact launch; uniform per wave anyway

  int rowA = base + (lane & 15);
  int koff = (lane >> 4) << 1;  // 0 for lanes 0-15, 2 for lanes 16-31
  const float* rp = nk + (size_t)rowA * EMB + koff;

  v8f c = {0.f, 0.f, 0.f, 0.f, 0.f, 0.f, 0.f, 0.f};
  v2f b; b.x = 1.0f; b.y = 1.0f;

#pragma unroll
  for (int k0 = 0; k0 < EMB; k0 += 4) {
    v2f a;
    a.x = rp[k0];
    a.y = rp[k0 + 1];
    a.x *= a.x;
    a.y *= a.y;
    // 8 args: (neg_a, A, neg_b, B, c_mod, C, reuse_a, reuse_b)
    c = __builtin_amdgcn_wmma_f32_16x16x4_f32(false, a, false, b, (short)0, c, false, false);
  }

  // C layout: VGPR r, lanes 0-15 -> M=r ; lanes 16-31 -> M=8+r (all N columns identical)
  if (lane == 0 || lane == 16) {
    int rb = base + ((lane >> 4) << 3);
#pragma unroll
    for (int r = 0; r < 8; ++r) {
      fac[rb + r] = EPS_F / fmaxf(sqrtf(c[r]), 1e-12f);
    }
  }
}

// ---------------- perturb in place + accumulate raw sum into out ----------------

__global__ void k_perturb_acc(float* __restrict__ ego, const float* __restrict__ fac,
                              const float* __restrict__ nk, float* __restrict__ acc, int nrows) {
  unsigned t = blockIdx.x * blockDim.x + threadIdx.x;
  unsigned row = t >> 4;
  if (row >= (unsigned)nrows) return;
  unsigned part = t & 15u;
  float f = fac[row];
  size_t off = (size_t)row * EMB + (size_t)part * 4;
  float4 e  = *(float4*)(ego + off);
  float4 nz = *(const float4*)(nk + off);
  e.x += (float)((e.x > 0.f) - (e.x < 0.f)) * nz.x * f;
  e.y += (float)((e.y > 0.f) - (e.y < 0.f)) * nz.y * f;
  e.z += (float)((e.z > 0.f) - (e.z < 0.f)) * nz.z * f;
  e.w += (float)((e.w > 0.f) - (e.w < 0.f)) * nz.w * f;
  *(float4*)(ego + off) = e;  // carried into next layer's SpMM
  float4 a = *(float4*)(acc + off);
  a.x += e.x; a.y += e.y; a.z += e.z; a.w += e.w;
  *(float4*)(acc + off) = a;
}

// ---------------- host orchestration ----------------

extern "C" void kernel_launch(void* const* d_in, const int* in_sizes, int n_in,
                              void* d_out, int out_size, void* d_ws, size_t ws_size,
                              hipStream_t stream) {
  const float* user_emb    = (const float*)d_in[0];
  const float* item_emb    = (const float*)d_in[1];
  const float* concept_emb = (const float*)d_in[2];
  const int*   adj_rows    = (const int*)d_in[3];
  const int*   adj_cols    = (const int*)d_in[4];
  const float* adj_vals    = (const float*)d_in[5];
  const int*   pre_rows    = (const int*)d_in[6];
  const int*   pre_cols    = (const int*)d_in[7];
  const float* pre_vals    = (const float*)d_in[8];
  const int*   ic_rows     = (const int*)d_in[9];
  const int*   ic_cols     = (const int*)d_in[10];
  const float* ic_vals     = (const float*)d_in[11];
  const float* noise       = (const float*)d_in[12];
  float* out = (float*)d_out;

  const int nnz_adj = in_sizes[3];
  const int nnz_pre = in_sizes[6];
  const int nnz_ic  = in_sizes[9];

  // workspace layout (floats): ego[N*64] | tmp[N*64] | ca[C*64] | cb[C*64] | fac[N]
  float* ego = (float*)d_ws;
  float* tmp = ego + (size_t)NTOTAL * EMB;
  float* ca  = tmp + (size_t)NTOTAL * EMB;
  float* cb  = ca  + (size_t)CONCEPT_NUM * EMB;
  float* fac = cb  + (size_t)CONCEPT_NUM * EMB;

  const int BLK = 256;
  const int n4_N = NTOTAL * EMB / 4;       // 2.4M float4
  const int n4_U = USER_NUM * EMB / 4;
  const int n4_I = ITEM_NUM * EMB / 4;
  const int n4_C = CONCEPT_NUM * EMB / 4;
  const int GS_BLOCKS = 2048;              // grid-stride block count for util kernels

  // init: ego = [user_emb ; item_emb], ca = concept_emb, out(acc) = 0
  k_copy4<<<GS_BLOCKS, BLK, 0, stream>>>((float4*)ego, (const float4*)user_emb, n4_U);
  k_copy4<<<GS_BLOCKS, BLK, 0, stream>>>((float4*)(ego + (size_t)USER_NUM * EMB),
                                         (const float4*)item_emb, n4_I);
  k_copy4<<<GS_BLOCKS, BLK, 0, stream>>>((float4*)ca, (const float4*)concept_emb, n4_C);
  k_zero4<<<GS_BLOCKS, BLK, 0, stream>>>((float4*)out, n4_N);

  for (int k = 0; k < N_LAYERS; ++k) {
    const float* nk = noise + (size_t)k * NTOTAL * EMB;

    // tmp = spmm(adj, ego)
    k_zero4<<<GS_BLOCKS, BLK, 0, stream>>>((float4*)tmp, n4_N);
    {
      long long thr = (long long)nnz_adj * 16;
      k_scatter<<<(unsigned)((thr + BLK - 1) / BLK), BLK, 0, stream>>>(
          adj_rows, adj_cols, adj_vals, ego, tmp, nnz_adj, 1.0f, 0);
    }

    // cb = ca + spmm(pre, ca)
    k_copy4<<<GS_BLOCKS, BLK, 0, stream>>>((float4*)cb, (const float4*)ca, n4_C);
    {
      long long thr = (long long)nnz_pre * 16;
      k_scatter<<<(unsigned)((thr + BLK - 1) / BLK), BLK, 0, stream>>>(
          pre_rows, pre_cols, pre_vals, ca, cb, nnz_pre, 1.0f, 0);
    }

    // tmp[item part] += 0.1 * spmm(ic, cb)
    {
      long long thr = (long long)nnz_ic * 16;
      k_scatter<<<(unsigned)((thr + BLK - 1) / BLK), BLK, 0, stream>>>(
          ic_rows, ic_cols, ic_vals, cb, tmp, nnz_ic, 0.1f, USER_NUM);
    }

    // fac[row] = EPS / ||nk[row]||  (WMMA f32 16x16x4 reduction)
    // exact wave launch: NTOTAL/16 = 9375 waves = 1875 blocks x 5 waves (blockDim 160)
    {
      const int NORM_BLK = 160;                     // 5 waves of 32
      int waves = NTOTAL / 16;                      // 9375
      int blocks = waves / (NORM_BLK / 32);         // 1875, exact
      k_norm_wmma<<<blocks, NORM_BLK, 0, stream>>>(nk, fac, NTOTAL);
    }

    // tmp += sign(tmp)*nk*fac ; out += tmp
    {
      long long thr = (long long)NTOTAL * 16;
      k_perturb_acc<<<(unsigned)((thr + BLK - 1) / BLK), BLK, 0, stream>>>(
          tmp, fac, nk, out, NTOTAL);
    }

    // swap for next layer
    float* t = ego; ego = tmp; tmp = t;
    float* t2 = ca; ca = cb; cb = t2;
  }

  // out = acc / N_LAYERS
  k_scale4<<<GS_BLOCKS, BLK, 0, stream>>>((float4*)out, 1.0f / (float)N_LAYERS, n4_N);
}